// BinarizeLayer_81810537055287
// MI455X (gfx1250) — compile-verified
//
#include <hip/hip_runtime.h>
#include <hip/hip_bf16.h>
#include <float.h>

// ---------------------------------------------------------------------------
// Otsu binarization for [32,128,2048,3] f32.
// ws layout (uint32 slots): [0]=min key, [1]=max key, [2..257]=hist, [258]=thresh bits
// ---------------------------------------------------------------------------

typedef float v2f __attribute__((ext_vector_type(2)));
typedef float v8f __attribute__((ext_vector_type(8)));

#define NBINS 256

__device__ __forceinline__ unsigned fkey(float f) {
  unsigned u = __float_as_uint(f);
  return (u & 0x80000000u) ? ~u : (u | 0x80000000u);
}
__device__ __forceinline__ float fdec(unsigned k) {
  unsigned u = (k & 0x80000000u) ? (k ^ 0x80000000u) : ~k;
  return __uint_as_float(u);
}
__device__ __forceinline__ float gray_of(float r, float g, float b) {
  return fmaf(0.2989f, r, fmaf(0.5870f, g, 0.1140f * b));
}

// ---------------------------------------------------------------------------
__global__ void otsu_init(unsigned* __restrict__ ws) {
  int t = threadIdx.x;
  if (t == 0) ws[0] = 0xFFFFFFFFu;   // min key identity (>= every key)
  if (t == 1) ws[1] = 0x00000000u;   // max key identity
  if (t == 2) ws[258] = 0u;
  if (t < NBINS) ws[2 + t] = 0u;
}

// ---------------------------------------------------------------------------
__global__ void gray_minmax(const float* __restrict__ in,
                            unsigned* __restrict__ ws, int npix) {
  int tid = blockIdx.x * blockDim.x + threadIdx.x;
  int p0 = tid * 4;
  float lmin = FLT_MAX, lmax = -FLT_MAX;
  if (p0 + 3 < npix) {
    const float4* ip = (const float4*)(in + (size_t)p0 * 3);
    float4 a = ip[0], b = ip[1], c = ip[2];
    float g0 = gray_of(a.x, a.y, a.z);
    float g1 = gray_of(a.w, b.x, b.y);
    float g2 = gray_of(b.z, b.w, c.x);
    float g3 = gray_of(c.y, c.z, c.w);
    lmin = fminf(fminf(g0, g1), fminf(g2, g3));
    lmax = fmaxf(fmaxf(g0, g1), fmaxf(g2, g3));
  } else {
    for (int p = p0; p < npix; ++p) {
      float g = gray_of(in[3 * (size_t)p], in[3 * (size_t)p + 1], in[3 * (size_t)p + 2]);
      lmin = fminf(lmin, g);
      lmax = fmaxf(lmax, g);
    }
  }
  __shared__ float smin[256];
  __shared__ float smax[256];
  smin[threadIdx.x] = lmin;
  smax[threadIdx.x] = lmax;
  __syncthreads();
  for (int s = 128; s > 0; s >>= 1) {
    if ((int)threadIdx.x < s) {
      smin[threadIdx.x] = fminf(smin[threadIdx.x], smin[threadIdx.x + s]);
      smax[threadIdx.x] = fmaxf(smax[threadIdx.x], smax[threadIdx.x + s]);
    }
    __syncthreads();
  }
  if (threadIdx.x == 0) {
    atomicMin(&ws[0], fkey(smin[0]));
    atomicMax(&ws[1], fkey(smax[0]));
  }
}

// ---------------------------------------------------------------------------
__global__ void gray_hist(const float* __restrict__ in,
                          unsigned* __restrict__ ws, int npix) {
  __shared__ unsigned shist[NBINS];
  shist[threadIdx.x] = 0u;  // blockDim.x == 256
  __syncthreads();

  float vmin = fdec(ws[0]);
  float vmax = fdec(ws[1]);
  float w = vmax - vmin;
  float scale = (w > 0.0f) ? (256.0f / w) : 0.0f;

  int tid = blockIdx.x * blockDim.x + threadIdx.x;
  int p0 = tid * 4;
  if (p0 + 3 < npix) {
    const float4* ip = (const float4*)(in + (size_t)p0 * 3);
    float4 a = ip[0], b = ip[1], c = ip[2];
    float g[4];
    g[0] = gray_of(a.x, a.y, a.z);
    g[1] = gray_of(a.w, b.x, b.y);
    g[2] = gray_of(b.z, b.w, c.x);
    g[3] = gray_of(c.y, c.z, c.w);
#pragma unroll
    for (int i = 0; i < 4; ++i) {
      int bin = (int)((g[i] - vmin) * scale);
      bin = bin < 0 ? 0 : (bin > 255 ? 255 : bin);  // last bin inclusive (jnp.histogram)
      atomicAdd(&shist[bin], 1u);
    }
  } else {
    for (int p = p0; p < npix; ++p) {
      float g = gray_of(in[3 * (size_t)p], in[3 * (size_t)p + 1], in[3 * (size_t)p + 2]);
      int bin = (int)((g - vmin) * scale);
      bin = bin < 0 ? 0 : (bin > 255 ? 255 : bin);
      atomicAdd(&shist[bin], 1u);
    }
  }
  __syncthreads();
  unsigned cnt = shist[threadIdx.x];
  if (cnt) atomicAdd(&ws[2 + threadIdx.x], cnt);
}

// ---------------------------------------------------------------------------
// Inclusive cumsum of 256 f32 values held in LDS, in place, using WMMA.
// View src as 16x16 row-major H (H[r][c] = src[16r+c]).
//   P = H * U   (U upper-triangular ones, K=16 via 4x v_wmma_f32_16x16x4_f32)
//   cum[16r+j] = P[r][j] + sum_{r'<r} P[r'][15]   (scalar block-prefix fixup)
// Wave32, EXEC all ones, uniform control flow (WMMA requirement).
// Layouts per CDNA5 ISA 7.12.2: A/B operand component c at lane-half hf holds
// K = 4*kb + 2*hf + c; D component v holds row (v + 8*hf), col = lane&15.
// ---------------------------------------------------------------------------
__device__ __forceinline__ void cumsum256_wmma(float* s, int lane) {
  const int hf = lane >> 4;
  const int lid = lane & 15;

  v8f acc = {0.f, 0.f, 0.f, 0.f, 0.f, 0.f, 0.f, 0.f};
#pragma unroll
  for (int kb = 0; kb < 4; ++kb) {
    const int k0 = 4 * kb + 2 * hf;
    v2f a, b;
    a.x = s[16 * lid + k0];          // A = H, row lid, cols k0, k0+1
    a.y = s[16 * lid + k0 + 1];
    b.x = (k0 <= lid) ? 1.0f : 0.0f; // B = U (upper-triangular ones), col lid
    b.y = (k0 + 1 <= lid) ? 1.0f : 0.0f;
    acc = __builtin_amdgcn_wmma_f32_16x16x4_f32(
        false, a, false, b, (short)0, acc, false, false);
  }
  __syncthreads();  // all reads of s complete before overwrite

  // Scatter D back: component v is P[v + 8*hf][lid]
#pragma unroll
  for (int v = 0; v < 8; ++v) {
    int m = v + 8 * hf;
    s[16 * m + lid] = acc[v];
  }
  __syncthreads();

  // Block-prefix fixup: bp[r] = sum of row totals of rows < r
  __shared__ float bp[16];
  if (lane < 16) {
    float t = 0.0f;
    for (int r = 0; r < lane; ++r) t += s[16 * r + 15];
    bp[lane] = t;
  }
  __syncthreads();
#pragma unroll
  for (int i = 0; i < 8; ++i) {
    int idx = 8 * lane + i;
    s[idx] = s[idx] + bp[idx >> 4];
  }
  __syncthreads();
}

__global__ __launch_bounds__(32) void otsu_kernel(unsigned* __restrict__ ws) {
  __shared__ float h[NBINS];   // becomes w1 (cumulative weights)
  __shared__ float hc[NBINS];  // becomes s1 (cumulative weighted centers)
  const int lane = threadIdx.x;

  float vmin = fdec(ws[0]);
  float vmax = fdec(ws[1]);
  float step = (vmax - vmin) * (1.0f / 256.0f);

#pragma unroll
  for (int i = 0; i < 8; ++i) {
    int bin = lane * 8 + i;
    float cnt = (float)ws[2 + bin];
    float ctr = vmin + ((float)bin + 0.5f) * step;
    h[bin] = cnt;
    hc[bin] = cnt * ctr;
  }
  __syncthreads();

  cumsum256_wmma(h, lane);
  cumsum256_wmma(hc, lane);

  if (lane == 0) {
    float total = h[255];
    float stotal = hc[255];
    float best = -1.0f;
    int bidx = 0;
    for (int t = 0; t < 255; ++t) {  // var12 over 255 split points, argmax = first max
      float W1 = h[t];
      float W2 = total - W1;             // w2[t+1]
      float M1 = hc[t] / fmaxf(W1, 1.0f);
      float M2 = (stotal - hc[t]) / fmaxf(W2, 1.0f);
      float d = M1 - M2;
      float v = W1 * W2 * d * d;
      if (v > best) { best = v; bidx = t; }
    }
    float thr = vmin + ((float)bidx + 0.5f) * step;
    ((float*)ws)[258] = thr;
  }
}

// ---------------------------------------------------------------------------
__global__ void binarize(const float* __restrict__ in, float* __restrict__ out,
                         const unsigned* __restrict__ ws, int npix) {
  float thr = __uint_as_float(ws[258]);
  int tid = blockIdx.x * blockDim.x + threadIdx.x;
  int p0 = tid * 4;
  if (p0 + 3 < npix) {
    const float4* ip = (const float4*)(in + (size_t)p0 * 3);
    float4 a = ip[0], b = ip[1], c = ip[2];
    float o0 = (gray_of(a.x, a.y, a.z) > thr) ? 1.0f : 0.0f;
    float o1 = (gray_of(a.w, b.x, b.y) > thr) ? 1.0f : 0.0f;
    float o2 = (gray_of(b.z, b.w, c.x) > thr) ? 1.0f : 0.0f;
    float o3 = (gray_of(c.y, c.z, c.w) > thr) ? 1.0f : 0.0f;
    float4* op = (float4*)(out + (size_t)p0 * 3);
    float4 r0, r1, r2;
    r0.x = o0; r0.y = o0; r0.z = o0; r0.w = o1;
    r1.x = o1; r1.y = o1; r1.z = o2; r1.w = o2;
    r2.x = o2; r2.y = o3; r2.z = o3; r2.w = o3;
    op[0] = r0; op[1] = r1; op[2] = r2;
  } else {
    for (int p = p0; p < npix; ++p) {
      float g = gray_of(in[3 * (size_t)p], in[3 * (size_t)p + 1], in[3 * (size_t)p + 2]);
      float o = (g > thr) ? 1.0f : 0.0f;
      out[3 * (size_t)p] = o;
      out[3 * (size_t)p + 1] = o;
      out[3 * (size_t)p + 2] = o;
    }
  }
}

// ---------------------------------------------------------------------------
extern "C" void kernel_launch(void* const* d_in, const int* in_sizes, int n_in,
                              void* d_out, int out_size, void* d_ws, size_t ws_size,
                              hipStream_t stream) {
  const float* in = (const float*)d_in[0];
  float* out = (float*)d_out;
  unsigned* ws = (unsigned*)d_ws;

  int nflt = in_sizes[0];
  int npix = nflt / 3;                       // 8,388,608
  int nthreads = (npix + 3) / 4;
  int blocks = (nthreads + 255) / 256;

  otsu_init<<<1, 256, 0, stream>>>(ws);
  gray_minmax<<<blocks, 256, 0, stream>>>(in, ws, npix);
  gray_hist<<<blocks, 256, 0, stream>>>(in, ws, npix);
  otsu_kernel<<<1, 32, 0, stream>>>(ws);
  binarize<<<blocks, 256, 0, stream>>>(in, out, ws, npix);
}